// SdfDPLoss_5669356834126
// MI455X (gfx1250) — compile-verified
//
#include <hip/hip_runtime.h>
#include <hip/hip_bf16.h>

typedef __attribute__((ext_vector_type(2))) float v2f;
typedef __attribute__((ext_vector_type(8))) float v8f;

#define BB 4
#define MM 8192
#define NN 6890
#define MT 2                           // M-tiles per wave (2 WMMAs per B-tile load)
#define NSPLIT 4                       // N-chunks (cross-wave split of the scan)
#define CHUNK 1728                     // ceil(NN/NSPLIT) rounded up to 16
#define TPB   ((MM / 16) / MT)         // m-tile-pairs per batch (256)
#define MIN2   (0.005f * 0.005f)       // MIN_DIST_THRESH^2
#define INFK   0x7F800000u             // +inf bit pattern: "no candidate" key

// A row m = (-2x,-2y,-2z, 1), B col n = (x,y,z,|s|^2)  =>  acc = |s|^2 - 2 c.s
// f = (|c|^2 - MIN2) + acc = d2 - MIN2.  Sort key = bits(f) as u32:
//   * valid d2 >= MIN2  -> f >= 0 -> u32-order == float-order
//   * d2 <  MIN2        -> f <  0 -> sign bit set -> never wins (excluded)
//   * invalid column    -> B K3 slot = +inf -> f = +inf -> ties init, never wins
// blockIdx.x encodes (b, tile-group) and blockIdx.y the n-chunk, so b/c/niter
// are block-uniform -> scalar loop control + SGPR base addressing.
__global__ __launch_bounds__(128) void nn_wmma_kernel(
    const float* __restrict__ cloth,        // [B,M,3]
    const int*   __restrict__ svalid,       // [B,N]
    const float* __restrict__ vt,           // [B,N,3]
    unsigned*    __restrict__ cand_k,       // [B*M*NSPLIT]
    int*         __restrict__ cand_n)       // [B*M*NSPLIT]
{
    const int lane  = threadIdx.x & 31;
    const int wave  = threadIdx.x >> 5;
    const int b     = blockIdx.x / (TPB / 4);              // block-uniform
    const int group = blockIdx.x % (TPB / 4);
    const int c     = blockIdx.y;                          // n-chunk, block-uniform
    const int tbase = (group * 4 + wave) * MT;             // first m-tile index

    const int nstart = c * CHUNK;
    const int nend   = (nstart + CHUNK < NN) ? (nstart + CHUNK) : NN;
    const int niter  = (nend - nstart) >> 4;               // full 16-column tiles

    const int half = lane >> 4;        // 0: lanes 0-15 (K=0,1), 1: lanes 16-31 (K=2,3)
    const int ln   = lane & 15;        // row (A) / column (B,C,D) index within tile
    const int k0   = half ? 2 : 0;
    const float INF = __uint_as_float(INFK);

    // ---------- A operands (loop invariant) ----------
    v2f      a[MT];
    float    c2m[MT][8];               // |c|^2 - MIN2 per accumulator row
    unsigned bestk[MT][8];
    int      bestn[MT][8];
#pragma unroll
    for (int t = 0; t < MT; ++t) {
        const int m0 = (tbase + t) * 16;
        const float* cb = cloth + ((size_t)b * MM + (size_t)(m0 + ln)) * 3;
        const float x0 = cb[k0];                 // x (lower) or z (upper)
        const float x1 = half ? 0.0f : cb[1];    // y (lower) or pad (upper)
        const float pp = x0 * x0 + x1 * x1;      // x^2+y^2 (lower) / z^2 (upper)
        const float c2 = pp + __shfl_xor(pp, 16, 32);
        a[t].x = -2.0f * x0;
        a[t].y = half ? 1.0f : -2.0f * x1;       // K3 slot = 1 in upper half
#pragma unroll
        for (int r = 0; r < 8; ++r) {
            c2m[t][r]   = __shfl(c2, r + 8 * half, 32) - MIN2;  // row m = r + 8*half
            bestk[t][r] = INFK;
            bestn[t][r] = 0;
        }
    }

    // SGPR bases (b, nstart block-uniform) + 32-bit per-lane offsets
    const float* vbase = vt     + (size_t)b * NN * 3 + (size_t)nstart * 3;
    const int*   valb  = svalid + (size_t)b * NN + nstart;
    int coff = ln * 3;                 // element offset into vbase
    int voff = ln;                     // element offset into valb

    // ---------- main loop: full 16-column tiles, scalar trip count ----------
    int n = nstart + ln;
    for (int it = 0; it < niter; ++it, coff += 48, voff += 16, n += 16) {
        const float b0 = vbase[coff + k0];       // x (lower) / z (upper)
        const float b1 = vbase[coff + k0 + 1];   // y (lower) / junk (upper, replaced)
        const bool vld = valb[voff] > 0;

        const float pp  = half ? b0 * b0 : (b0 * b0 + b1 * b1);
        const float oth = __shfl_xor(pp, 16, 32);
        const float s2  = vld ? (pp + oth) : INF;   // invalid column -> +inf
        v2f bm;
        bm.x = b0;
        bm.y = half ? s2 : b1;                   // upper K3 slot carries |s|^2

#pragma unroll
        for (int t = 0; t < MT; ++t) {
            v8f acc = {};
            acc = __builtin_amdgcn_wmma_f32_16x16x4_f32(
                false, a[t], false, bm, (short)0, acc, false, false);
#pragma unroll
            for (int r = 0; r < 8; ++r) {
                const unsigned k = __float_as_uint(c2m[t][r] + acc[r]);
                if (k < bestk[t][r]) { bestk[t][r] = k; bestn[t][r] = n; }
            }
        }
    }

    // ---------- tail tile (only last chunk), guarded ----------
    if (nstart + (niter << 4) < nend) {
        const bool inb = (n < nend);
        v2f bm; bm.x = 0.0f; bm.y = 0.0f;
        bool vld = false;
        if (inb) {
            bm.x = vbase[coff + k0];
            bm.y = half ? 0.0f : vbase[coff + 1];
            vld  = (valb[voff] > 0);
        }
        const float pp  = half ? bm.x * bm.x : (bm.x * bm.x + bm.y * bm.y);
        const float oth = __shfl_xor(pp, 16, 32);
        if (half) bm.y = vld ? (pp + oth) : INF;

#pragma unroll
        for (int t = 0; t < MT; ++t) {
            v8f acc = {};
            acc = __builtin_amdgcn_wmma_f32_16x16x4_f32(
                false, a[t], false, bm, (short)0, acc, false, false);
#pragma unroll
            for (int r = 0; r < 8; ++r) {
                const unsigned k = __float_as_uint(c2m[t][r] + acc[r]);
                if (k < bestk[t][r]) { bestk[t][r] = k; bestn[t][r] = n; }
            }
        }
    }

    // ---------- cross-lane min+argmin (16 columns live in the lanes of a half) ----------
#pragma unroll
    for (int t = 0; t < MT; ++t) {
#pragma unroll
        for (int r = 0; r < 8; ++r) {
#pragma unroll
            for (int off = 1; off < 16; off <<= 1) {
                unsigned ok = (unsigned)__shfl_xor((int)bestk[t][r], off, 32);
                int      on = __shfl_xor(bestn[t][r], off, 32);
                if (ok < bestk[t][r] || (ok == bestk[t][r] && on < bestn[t][r])) {
                    bestk[t][r] = ok; bestn[t][r] = on;
                }
            }
        }
    }

    // lanes 0 and 16 hold reduced results for rows r and r+8 of each tile
    if (ln == 0) {
#pragma unroll
        for (int t = 0; t < MT; ++t) {
            const int m0 = (tbase + t) * 16;
#pragma unroll
            for (int r = 0; r < 8; ++r) {
                const int m = m0 + r + 8 * half;
                const size_t o = ((size_t)b * MM + m) * NSPLIT + c;
                cand_k[o] = bestk[t][r];
                cand_n[o] = bestn[t][r];
            }
        }
    }
}

// Merge NSPLIT candidates per m (u32 key order + index tie-break => exact
// first-occurrence argmin), then gather/match/loss and per-batch mean*any.
__global__ __launch_bounds__(256) void merge_reduce_kernel(
    const unsigned* __restrict__ cand_k,
    const int*      __restrict__ cand_n,
    const float*    __restrict__ sdf,          // [B,M]
    const int*      __restrict__ sidx,         // [B,N]
    const int*      __restrict__ cidx,         // [n_cidx]
    const float*    __restrict__ sdf_thresh,   // [1]
    const float*    __restrict__ dist_thresh,  // [1]
    int n_cidx,
    float* __restrict__ out)                   // [B]
{
    const int b = blockIdx.x;
    const float dt  = dist_thresh[0];
    const float dt2 = dt * dt;
    const float st  = sdf_thresh[0];

    __shared__ float ssum[256];
    __shared__ int   smat[256];

    float sum = 0.0f;
    int   any = 0;
    for (int m = threadIdx.x; m < MM; m += 256) {
        const size_t base = ((size_t)b * MM + m) * NSPLIT;
        unsigned bk = cand_k[base];
        int      bn = cand_n[base];
#pragma unroll
        for (int cc = 1; cc < NSPLIT; ++cc) {
            const unsigned k = cand_k[base + cc];
            const int      n = cand_n[base + cc];
            if (k < bk || (k == bk && n < bn)) { bk = k; bn = n; }
        }
        const int tg = sidx[(size_t)b * NN + bn];
        bool match = false;
        for (int j = 0; j < n_cidx; ++j) match = match || (tg == cidx[j]);
        const float d2 = __uint_as_float(bk) + MIN2;   // inf stays inf
        const bool near_ = d2 < dt2;
        const float s = sdf[(size_t)b * MM + m];
        if (near_) sum += match ? fabsf(s) : fabsf(s - st);
        any |= match ? 1 : 0;
    }
    ssum[threadIdx.x] = sum;
    smat[threadIdx.x] = any;
    __syncthreads();
    for (int off = 128; off > 0; off >>= 1) {
        if (threadIdx.x < off) {
            ssum[threadIdx.x] += ssum[threadIdx.x + off];
            smat[threadIdx.x] |= smat[threadIdx.x + off];
        }
        __syncthreads();
    }
    if (threadIdx.x == 0)
        out[b] = (ssum[0] / (float)MM) * (smat[0] ? 1.0f : 0.0f);
}

extern "C" void kernel_launch(void* const* d_in, const int* in_sizes, int n_in,
                              void* d_out, int out_size, void* d_ws, size_t ws_size,
                              hipStream_t stream) {
    const float* sdf   = (const float*)d_in[0];
    const float* cloth = (const float*)d_in[1];
    const int*   sidx  = (const int*)  d_in[2];
    const int*   sval  = (const int*)  d_in[3];
    const int*   cidx  = (const int*)  d_in[4];
    const float* sth   = (const float*)d_in[5];
    const float* dth   = (const float*)d_in[6];
    const float* vt    = (const float*)d_in[7];
    const int n_cidx   = in_sizes[4];

    unsigned* cand_k = (unsigned*)d_ws;
    int*      cand_n = (int*)((char*)d_ws + (size_t)BB * MM * NSPLIT * sizeof(unsigned));

    dim3 grid(BB * (TPB / 4), NSPLIT);               // 256 x 4 blocks, 4 waves each
    nn_wmma_kernel<<<grid, 128, 0, stream>>>(cloth, sval, vt, cand_k, cand_n);

    merge_reduce_kernel<<<BB, 256, 0, stream>>>(
        cand_k, cand_n, sdf, sidx, cidx, sth, dth, n_cidx, (float*)d_out);
}